// GlobalSelfAttention3D_18485539242790
// MI455X (gfx1250) — compile-verified
//
#include <hip/hip_runtime.h>

typedef __attribute__((ext_vector_type(2))) float v2f;
typedef __attribute__((ext_vector_type(8))) float v8f;

#define N_TOK   1728        // 12*12*12 tokens
#define CDIM    256
#define NHEADS  8
#define HD      32
#define NT      108         // N_TOK / 16
#define SCALE_  0.1767766952966369f   // 32^-0.5

// D(16x16,f32) = A(16x4,f32) * B(4x16,f32) + C  -> v_wmma_f32_16x16x4_f32
__device__ __forceinline__ v8f wmma4(v2f a, v2f b, v8f c) {
  return __builtin_amdgcn_wmma_f32_16x16x4_f32(
      /*neg_a=*/false, a, /*neg_b=*/false, b,
      /*c_mod=*/(short)0, c, /*reuse_a=*/false, /*reuse_b=*/false);
}

// ---------------------------------------------------------------------------
// Kernel 1: QKV = X^T (N x C) @ Wqkv^T (C x 768); scatter into Q/K/V [h][n][d]
// x is stored channel-major: x[c*N + n].  wqkv[j*C + c].
// One 16x16 output tile per wave; 108*48 = 5184 tiles; 8 waves/block.
// ---------------------------------------------------------------------------
__global__ __launch_bounds__(256)
void gsa3d_qkv_kernel(const float* __restrict__ x,
                      const float* __restrict__ wqkv,
                      float* __restrict__ Q,
                      float* __restrict__ K,
                      float* __restrict__ V) {
  const int wid  = blockIdx.x * 8 + (threadIdx.x >> 5);
  const int lane = threadIdx.x & 31;
  const int l16  = lane & 15;
  const int kh   = lane >> 4;          // which K-half (k 0..1 vs 2..3)
  const int tn   = wid % 48;           // tile along j (768/16)
  const int tm   = wid / 48;           // tile along n (1728/16)
  const int row  = tm * 16 + l16;      // A row (token index)
  const int col  = tn * 16 + l16;      // B col (j index)

  v8f acc = {};
  for (int k0 = 0; k0 < CDIM; k0 += 4) {
    const int ka = k0 + 2 * kh;
    v2f a, b;
    a.x = x[(ka + 0) * N_TOK + row];     // A[m][k] = x[k*N + n]
    a.y = x[(ka + 1) * N_TOK + row];
    b.x = wqkv[col * CDIM + ka + 0];     // B[k][j] = wqkv[j*C + k]
    b.y = wqkv[col * CDIM + ka + 1];
    acc = wmma4(a, b, acc);
  }

  // j = col -> (s in {q,k,v}, head, d)
  const int s  = col >> 8;
  const int hh = (col >> 5) & 7;
  const int d  = col & 31;
  float* dst = (s == 0) ? Q : ((s == 1) ? K : V);
  dst += hh * (N_TOK * HD) + d;
#pragma unroll
  for (int r = 0; r < 8; ++r) {
    const int n = tm * 16 + r + 8 * kh;  // D row mapping
    dst[n * HD] = acc[r];
  }
}

// ---------------------------------------------------------------------------
// Kernel 2: streaming (flash) attention per (head, 16-row Q block).
// S = (Q K^T) * scale, clip [-10,10], online softmax, O = P V, O /= l.
// One wave per task; 8 tasks/block; P tile staged through 1KB LDS/wave to
// convert D-layout -> A-layout for the P*V WMMA.
// ---------------------------------------------------------------------------
__global__ __launch_bounds__(256)
void gsa3d_attn_kernel(const float* __restrict__ Q,
                       const float* __restrict__ K,
                       const float* __restrict__ V,
                       float* __restrict__ AO) {
  __shared__ float ldsP[8 * 256];
  const int wslot = threadIdx.x >> 5;
  const int wid   = blockIdx.x * 8 + wslot;
  const int lane  = threadIdx.x & 31;
  const int l16   = lane & 15;
  const int kh    = lane >> 4;
  const int h     = wid / NT;
  const int tq    = wid % NT;

  const float* Qh = Q + h * (N_TOK * HD);
  const float* Kh = K + h * (N_TOK * HD);
  const float* Vh = V + h * (N_TOK * HD);
  float* P = ldsP + wslot * 256;

  // Q row-block A-fragments for all 8 K-steps (K=32), kept in registers.
  v2f aq[8];
  {
    const int row = tq * 16 + l16;
#pragma unroll
    for (int t = 0; t < 8; ++t) {
      const int ka = 4 * t + 2 * kh;
      aq[t].x = Qh[row * HD + ka + 0];
      aq[t].y = Qh[row * HD + ka + 1];
    }
  }

  float m_i[8], l_i[8];
#pragma unroll
  for (int r = 0; r < 8; ++r) { m_i[r] = -1.0e30f; l_i[r] = 0.0f; }
  v8f o0 = {};   // O columns 0..15
  v8f o1 = {};   // O columns 16..31

  for (int kb = 0; kb < NT; ++kb) {
    // S tile = Q_block (16x32) * K_block^T (32x16)
    v8f s = {};
#pragma unroll
    for (int t = 0; t < 8; ++t) {
      const int kd = 4 * t + 2 * kh;
      v2f b;                                   // B[k][m] = Kh[m*HD + k]
      b.x = Kh[(kb * 16 + l16) * HD + kd + 0];
      b.y = Kh[(kb * 16 + l16) * HD + kd + 1];
      s = wmma4(aq[t], b, s);
    }

    float p[8], alpha[8];
#pragma unroll
    for (int r = 0; r < 8; ++r) {
      float sv = s[r] * SCALE_;
      sv = fminf(fmaxf(sv, -10.0f), 10.0f);
      // row max across the 16-lane column group (khalf bit preserved)
      float mx = sv;
      mx = fmaxf(mx, __shfl_xor(mx, 1, 32));
      mx = fmaxf(mx, __shfl_xor(mx, 2, 32));
      mx = fmaxf(mx, __shfl_xor(mx, 4, 32));
      mx = fmaxf(mx, __shfl_xor(mx, 8, 32));
      const float mnew = fmaxf(m_i[r], mx);
      alpha[r] = __expf(m_i[r] - mnew);
      p[r]     = __expf(sv - mnew);
      float rs = p[r];
      rs += __shfl_xor(rs, 1, 32);
      rs += __shfl_xor(rs, 2, 32);
      rs += __shfl_xor(rs, 4, 32);
      rs += __shfl_xor(rs, 8, 32);
      l_i[r] = l_i[r] * alpha[r] + rs;
      m_i[r] = mnew;
    }

    // rescale accumulators, stage P (D-layout) into LDS
#pragma unroll
    for (int r = 0; r < 8; ++r) {
      o0[r] *= alpha[r];
      o1[r] *= alpha[r];
      P[(r + 8 * kh) * 16 + l16] = p[r];       // P[row][col]
    }

    // O += P (16x16) * V_block (16x32); reload P in A-layout from LDS
#pragma unroll
    for (int t = 0; t < 4; ++t) {
      const int kp = 4 * t + 2 * kh;
      v2f ap, b0, b1;
      ap.x = P[l16 * 16 + kp + 0];
      ap.y = P[l16 * 16 + kp + 1];
      const float* vr0 = Vh + (kb * 16 + kp + 0) * HD;
      const float* vr1 = Vh + (kb * 16 + kp + 1) * HD;
      b0.x = vr0[l16];        b0.y = vr1[l16];
      b1.x = vr0[16 + l16];   b1.y = vr1[16 + l16];
      o0 = wmma4(ap, b0, o0);
      o1 = wmma4(ap, b1, o1);
    }
  }

  // normalize and write AO[n][h*32 + d] (row-major N x C)
#pragma unroll
  for (int r = 0; r < 8; ++r) {
    const int n   = tq * 16 + r + 8 * kh;
    const float inv = 1.0f / l_i[r];
    AO[n * CDIM + h * HD + l16]      = o0[r] * inv;
    AO[n * CDIM + h * HD + 16 + l16] = o1[r] * inv;
  }
}

// ---------------------------------------------------------------------------
// Kernel 3: OUT = AO (N x C) @ Wproj^T (C x C), stored transposed as (C, N).
// 108 * 16 = 1728 tiles; 8 waves/block.
// ---------------------------------------------------------------------------
__global__ __launch_bounds__(256)
void gsa3d_proj_kernel(const float* __restrict__ AO,
                       const float* __restrict__ wproj,
                       float* __restrict__ out) {
  const int wid  = blockIdx.x * 8 + (threadIdx.x >> 5);
  const int lane = threadIdx.x & 31;
  const int l16  = lane & 15;
  const int kh   = lane >> 4;
  const int tm   = wid / 16;           // token tile
  const int tn   = wid % 16;           // output-channel tile
  const int row  = tm * 16 + l16;
  const int col  = tn * 16 + l16;

  v8f acc = {};
  for (int k0 = 0; k0 < CDIM; k0 += 4) {
    const int ka = k0 + 2 * kh;
    v2f a, b;
    a.x = AO[row * CDIM + ka + 0];
    a.y = AO[row * CDIM + ka + 1];
    b.x = wproj[col * CDIM + ka + 0];  // B[k][j] = wproj[j*C + k]
    b.y = wproj[col * CDIM + ka + 1];
    acc = wmma4(a, b, acc);
  }
#pragma unroll
  for (int r = 0; r < 8; ++r) {
    const int n = tm * 16 + r + 8 * kh;
    out[col * N_TOK + n] = acc[r];     // output is (C, N) channel-major
  }
}

// ---------------------------------------------------------------------------
extern "C" void kernel_launch(void* const* d_in, const int* in_sizes, int n_in,
                              void* d_out, int out_size, void* d_ws, size_t ws_size,
                              hipStream_t stream) {
  (void)in_sizes; (void)n_in; (void)out_size; (void)ws_size;
  const float* x     = (const float*)d_in[0];   // (1,256,12,12,12)
  const float* wqkv  = (const float*)d_in[1];   // (768,256)
  const float* wproj = (const float*)d_in[2];   // (256,256)
  float* out = (float*)d_out;                   // (1,256,12,12,12)

  float* ws = (float*)d_ws;
  float* Q  = ws;                                // [8][1728][32]
  float* K  = Q + NHEADS * N_TOK * HD;
  float* V  = K + NHEADS * N_TOK * HD;
  float* AO = V + NHEADS * N_TOK * HD;           // [1728][256]

  gsa3d_qkv_kernel <<<648, 256, 0, stream>>>(x, wqkv, Q, K, V);   // 5184 tiles
  gsa3d_attn_kernel<<<108, 256, 0, stream>>>(Q, K, V, AO);        //  864 tasks
  gsa3d_proj_kernel<<<216, 256, 0, stream>>>(AO, wproj, out);     // 1728 tiles
}